// GCN_5987184411137
// MI455X (gfx1250) — compile-verified
//
#include <hip/hip_runtime.h>
#include <hip/hip_bf16.h>

typedef __attribute__((ext_vector_type(2))) float v2f;
typedef __attribute__((ext_vector_type(8))) float v8f;

#define IN_CH   65
#define HID_CH  64
#define KPAD    68          // 65 padded up to a multiple of 4
#define KTOT    136         // [aggr_pad | x_pad]

__device__ __forceinline__ void atomAddF(float* p, float v) {
#if defined(__gfx1250__)
    // Guaranteed native no-return fp32 atomic (VGLOBAL GLOBAL_ATOMIC_ADD_F32),
    // tracked on STOREcnt; s_endpgm's implicit wait-idle covers completion.
    asm volatile("global_atomic_add_f32 %0, %1, off" : : "v"(p), "v"(v) : "memory");
#else
    __hip_atomic_fetch_add(p, v, __ATOMIC_RELAXED, __HIP_MEMORY_SCOPE_AGENT);
#endif
}

// K1: Apad[n][k]: cols [0,68)=0 (scatter target), [68,133)=x[n], [133,136)=0
__global__ void gc_init_apad(const float* __restrict__ x, float* __restrict__ Apad, int nNodes) {
    int n = blockIdx.x;
    int k = threadIdx.x;
    if (n >= nNodes || k >= KTOT) return;
    float v = 0.0f;
    if (k >= KPAD && k < KPAD + IN_CH) v = x[(size_t)n * IN_CH + (k - KPAD)];
    Apad[(size_t)n * KTOT + k] = v;
}

// K2: Bmat[136][64]: rows [0,65)=W_rel^T, [68,133)=W_root^T, else 0
__global__ void gc_init_bmat(const float* __restrict__ Wrel, const float* __restrict__ Wroot,
                             float* __restrict__ Bmat) {
    int idx = blockIdx.x * blockDim.x + threadIdx.x;
    if (idx >= KTOT * HID_CH) return;
    int k = idx >> 6;
    int n = idx & 63;
    float v = 0.0f;
    if (k < IN_CH)                          v = Wrel[n * IN_CH + k];
    else if (k >= KPAD && k < KPAD + IN_CH) v = Wroot[n * IN_CH + (k - KPAD)];
    Bmat[idx] = v;
}

// K3: per-edge scatter: Apad[dst][0..64] += w * x[src][0..64]  (L2-resident atomics)
__global__ void gc_scatter(const float* __restrict__ x, const int* __restrict__ ei,
                           const float* __restrict__ ew, float* __restrict__ Apad,
                           int nEdges) {
    const int lane = threadIdx.x & 31;
    const long wid = (long)((blockIdx.x * blockDim.x + threadIdx.x) >> 5);
    long e0 = wid * 4;
    for (int i = 0; i < 4; ++i) {
        long e = e0 + i;
        if (e >= nEdges) return;
        int   src = ei[e];
        int   dst = ei[(long)nEdges + e];
        float wt  = ew[e];
        const float* xr = x    + (size_t)src * IN_CH;
        float*       ar = Apad + (size_t)dst * KTOT;
        atomAddF(ar + lane,      wt * xr[lane]);
        atomAddF(ar + 32 + lane, wt * xr[32 + lane]);
        if (lane == 0) atomAddF(ar + 64, wt * xr[64]);
    }
}

// K4: out = Apad(100000x136) @ Bmat(136x64) + b_rel, via V_WMMA_F32_16X16X4_F32.
// One wave -> 16 rows x 64 cols (4 accumulator tiles). blockDim = 128 (4 waves).
__global__ void gc_gemm(const float* __restrict__ Apad, const float* __restrict__ Bmat,
                        const float* __restrict__ brel, float* __restrict__ out, int mTiles) {
    const int lane = threadIdx.x & 31;
    const int wv   = threadIdx.x >> 5;
    const int mt   = blockIdx.x * 4 + wv;
    if (mt >= mTiles) return;            // wave-uniform: EXEC all-ones for live waves
    const int r  = lane & 15;            // M row (A) / N col (B,C)
    const int hi = lane >> 4;            // K sub-pair select
    const float* arow = Apad + (size_t)(mt * 16 + r) * KTOT;
    v8f c0 = {}, c1 = {}, c2 = {}, c3 = {};
    for (int kc = 0; kc < KTOT; kc += 4) {
        // A-frag: VGPR0 = K+{0|2}, VGPR1 = K+{1|3} -> consecutive memory -> one b64 load
        v2f a = *(const v2f*)(arow + kc + hi * 2);
        const float* bp = Bmat + (size_t)(kc + hi * 2) * HID_CH + r;
        v2f b0, b1, b2, b3;
        b0.x = bp[0];  b0.y = bp[HID_CH + 0];
        b1.x = bp[16]; b1.y = bp[HID_CH + 16];
        b2.x = bp[32]; b2.y = bp[HID_CH + 32];
        b3.x = bp[48]; b3.y = bp[HID_CH + 48];
        c0 = __builtin_amdgcn_wmma_f32_16x16x4_f32(false, a, false, b0, (short)0, c0, false, false);
        c1 = __builtin_amdgcn_wmma_f32_16x16x4_f32(false, a, false, b1, (short)0, c1, false, false);
        c2 = __builtin_amdgcn_wmma_f32_16x16x4_f32(false, a, false, b2, (short)0, c2, false, false);
        c3 = __builtin_amdgcn_wmma_f32_16x16x4_f32(false, a, false, b3, (short)0, c3, false, false);
    }
    float bias0 = brel[r], bias1 = brel[16 + r], bias2 = brel[32 + r], bias3 = brel[48 + r];
    for (int i = 0; i < 8; ++i) {
        int m = mt * 16 + i + hi * 8;     // C layout: VGPR i -> M=i (lanes 0-15) / M=8+i (16-31)
        float* o = out + (size_t)m * HID_CH + r;
        o[0]  = c0[i] + bias0;
        o[16] = c1[i] + bias1;
        o[32] = c2[i] + bias2;
        o[48] = c3[i] + bias3;
    }
}

extern "C" void kernel_launch(void* const* d_in, const int* in_sizes, int n_in,
                              void* d_out, int out_size, void* d_ws, size_t ws_size,
                              hipStream_t stream) {
    const float* x     = (const float*)d_in[0];
    const int*   ei    = (const int*)  d_in[1];   // [2][E]: src row then dst row
    const float* ew    = (const float*)d_in[2];
    const float* Wrel  = (const float*)d_in[3];
    const float* brel  = (const float*)d_in[4];
    const float* Wroot = (const float*)d_in[5];
    float* out = (float*)d_out;

    const int nNodes = in_sizes[0] / IN_CH;  // 100000 (multiple of 16)
    const int nEdges = in_sizes[2];          // 3200000

    float* Apad = (float*)d_ws;                         // nNodes*136 floats (54.4 MB)
    float* Bmat = Apad + (size_t)nNodes * KTOT;         // +136*64 floats

    gc_init_apad<<<nNodes, 160, 0, stream>>>(x, Apad, nNodes);
    gc_init_bmat<<<(KTOT * HID_CH + 255) / 256, 256, 0, stream>>>(Wrel, Wroot, Bmat);

    long waves  = ((long)nEdges + 3) / 4;               // 4 edges per wave
    long blocks = (waves + 7) / 8;                      // 8 waves per block (256 thr)
    gc_scatter<<<dim3((unsigned)blocks), 256, 0, stream>>>(x, ei, ew, Apad, nEdges);

    int mTiles = nNodes / 16;                           // 6250
    gc_gemm<<<(mTiles + 3) / 4, 128, 0, stream>>>(Apad, Bmat, brel, out, mTiles);
}